// InputGradientExplainer_25890062860800
// MI455X (gfx1250) — compile-verified
//
#include <hip/hip_runtime.h>
#include <cstdint>
#include <cstddef>

// ---------------- problem constants ----------------
#define NN 1024   // stocks
#define DD 64     // feature dim
#define RR 32     // relations
static constexpr float NEGF = -1e14f;

typedef float f4  __attribute__((ext_vector_type(4)));
typedef float v2f __attribute__((ext_vector_type(2)));
typedef float v8f __attribute__((ext_vector_type(8)));
typedef int   v4i __attribute__((ext_vector_type(4)));

typedef __attribute__((address_space(3))) void* as3p;

#ifndef __has_builtin
#define __has_builtin(x) 0
#endif

// ---------------- CDNA5 async global->LDS copy (16B granule) ----------------
__device__ __forceinline__ void cp16_async(void* lds_dst, const void* gsrc) {
#if __has_builtin(__builtin_amdgcn_global_load_async_to_lds_b128)
  // signature (per hipcc diagnostic): takes pointers to 16B int vectors
  __builtin_amdgcn_global_load_async_to_lds_b128((v4i*)gsrc, (v4i*)lds_dst, 0, 0);
#else
  unsigned ldsa = (unsigned)(size_t)(as3p)lds_dst;
  asm volatile("global_load_async_to_lds_b128 %0, %1, off"
               :: "v"(ldsa), "v"(gsrc) : "memory");
#endif
}

__device__ __forceinline__ void wait_async0() {
#if __has_builtin(__builtin_amdgcn_s_wait_asynccnt)
  __builtin_amdgcn_s_wait_asynccnt(0);
#else
  asm volatile("s_wait_asynccnt 0" ::: "memory");
#endif
}

// ---------------- LDS tile geometry (bank-conflict-free: strides == 4 mod 64) --
#define LG_STRIDE 1028            // grad tile: [m=16][1024 floats + 4 pad]
#define LR_MSTR   68              // rs tile inner stride per m (64 + 4 pad)
#define LR_STRIDE (16 * LR_MSTR)  // rs tile: [jl=16][16*68]
#define SMEM_FLOATS (16 * LG_STRIDE + 16 * LR_STRIDE + 4)

// ---------------- k0: zero accumulators ----------------
__global__ void k0_init(float* __restrict__ acc) {
  if (threadIdx.x < 4) acc[threadIdx.x] = 0.0f;
}

// ---------------- k1: ewm via WMMA diag + async LDS staging ----------------
// ewm[i,j] = sum_d grad[i,j,d] * rs[j,i,d]; also accumulate {cnt!=0, sum, sumsq}
__global__ __launch_bounds__(256) void k1_ewm_wmma(
    const float* __restrict__ rs, const float* __restrict__ grad,
    float* __restrict__ ewm, float* __restrict__ acc) {
  extern __shared__ float smem[];
  float* lds_g = smem;                                  // grad[i0+m][j0+jl][d]
  float* lds_r = smem + 16 * LG_STRIDE;                 // rs[j0+jl][i0+m][d]
  float* s_acc = smem + 16 * LG_STRIDE + 16 * LR_STRIDE;

  const int t  = threadIdx.x;
  const int tj = blockIdx.x & 63;
  const int ti = blockIdx.x >> 6;
  const int i0 = ti * 16, j0 = tj * 16;

  if (t < 3) s_acc[t] = 0.0f;

  // Stage 128KB: 16 rows x 4KB per array; thread t moves granule t of each row.
  const float* gbase = grad + ((size_t)i0 * NN + j0) * DD;
  const float* rbase = rs   + ((size_t)j0 * NN + i0) * DD;
#pragma unroll
  for (int r = 0; r < 16; ++r) {
    cp16_async(lds_g + (size_t)r * LG_STRIDE + t * 4,
               gbase + (size_t)r * NN * DD + t * 4);
    cp16_async(lds_r + (size_t)r * LR_STRIDE + (t >> 4) * LR_MSTR + (t & 15) * 4,
               rbase + (size_t)r * NN * DD + t * 4);
  }
  wait_async0();
  __syncthreads();

  const int lane = t & 31;
  const int wave = t >> 5;
  const int lm   = lane & 15;
  const int koff = (lane >> 4) << 1;  // lanes 0-15 -> K {0,1}; 16-31 -> K {2,3}

  float lsum = 0.0f, lsq = 0.0f, lcnt = 0.0f;

#pragma unroll
  for (int jj = 0; jj < 2; ++jj) {
    const int jl = wave * 2 + jj;     // this wave's j column
    const float* pa = lds_g + (size_t)lm * LG_STRIDE + jl * DD;      // A: grad[i0+m, j, d]
    const float* pb = lds_r + (size_t)jl * LR_STRIDE + lm * LR_MSTR; // B: rs[j, i0+n, d]

#if __has_builtin(__builtin_amdgcn_wmma_f32_16x16x4_f32)
    v8f c = {0.f, 0.f, 0.f, 0.f, 0.f, 0.f, 0.f, 0.f};
#pragma unroll
    for (int kk = 0; kk < 16; ++kk) {  // K = 64 in chains of 4, full f32
      v2f a = { pa[4 * kk + koff], pa[4 * kk + koff + 1] };
      v2f b = { pb[4 * kk + koff], pb[4 * kk + koff + 1] };
      c = __builtin_amdgcn_wmma_f32_16x16x4_f32(false, a, false, b,
                                                (short)0, c, false, false);
    }
    // Diagonal extraction: C vgpr q holds (M=q, N=lane) for lane<16 and
    // (M=q+8, N=lane-16) for lane>=16. Diag r<8 -> lane r, c[r]; r>=8 -> lane r+16, c[r-8].
    const int  sel   = (lane < 16) ? lane : (lane - 16);
    const bool valid = sel < 8;
    const int  row   = (lane < 16) ? sel : sel + 8;
    float v = c[0];
#pragma unroll
    for (int q = 1; q < 8; ++q) v = (sel == q) ? c[q] : v;
#else
    // VALU fallback: lanes 0..15 each compute one row's dot product.
    const bool valid = lane < 16;
    const int  row   = lm;
    float v = 0.0f;
#pragma unroll
    for (int d = 0; d < DD; ++d) v = fmaf(pa[d], pb[d], v);
#endif
    if (valid) {
      ewm[(size_t)(i0 + row) * NN + (j0 + jl)] = v;
      lsum += v;
      lsq  += v * v;
      lcnt += (v != 0.0f) ? 1.0f : 0.0f;
    }
  }

  atomicAdd(&s_acc[0], lcnt);
  atomicAdd(&s_acc[1], lsum);
  atomicAdd(&s_acc[2], lsq);
  __syncthreads();
  if (t == 0) {
    atomicAdd(&acc[0], s_acc[0]);
    atomicAdd(&acc[1], s_acc[1]);
    atomicAdd(&acc[2], s_acc[2]);
  }
}

// ---------------- k2: mean and 1/std (unbiased) from accumulators ----------------
__global__ void k2_stats(const float* __restrict__ acc, float* __restrict__ stat) {
  const float cnt = acc[0], sum = acc[1], sq = acc[2];
  const float mean = sum / cnt;
  const float var  = (sq - sum * sum / cnt) / (cnt - 1.0f);
  stat[0] = mean;
  stat[1] = rsqrtf(var);  // 1/std
}

// ---------------- k3: masked softmax over R, sigmoid scale, output ----------------
__global__ __launch_bounds__(256) void k3_softmax_scale(
    const float* __restrict__ rm, const float* __restrict__ rmg,
    const float* __restrict__ ewm, const float* __restrict__ stat,
    float* __restrict__ out) {
  const size_t pix = (size_t)blockIdx.x * 256 + threadIdx.x;  // (i*N + j)
  const float mean = stat[0], rstd = stat[1];

  const float e = ewm[pix];
  const float normed = (e != 0.0f) ? (e - mean) * rstd : e;
  float sig = 1.0f / (1.0f + __expf(-normed));
  if (e == 0.0f) sig = 0.0f;

  const f4* rm4 = (const f4*)(rm  + pix * RR);
  const f4* rg4 = (const f4*)(rmg + pix * RR);

  f4 M[8], G[8];
  float mx = -3.402823466e38f, mn = 3.402823466e38f;
#pragma unroll
  for (int q = 0; q < 8; ++q) {
    M[q] = __builtin_nontemporal_load(rm4 + q);
    f4 g = __builtin_nontemporal_load(rg4 + q);
    g = g * M[q];
    G[q] = g;
#pragma unroll
    for (int c = 0; c < 4; ++c) {
      mx = fmaxf(mx, g[c]);
      mn = fminf(mn, g[c]);
    }
  }

  const float inv = 2.0f / (mx - mn);
  float rmax = -3.402823466e38f;
  f4 RES[8];
#pragma unroll
  for (int q = 0; q < 8; ++q) {
#pragma unroll
    for (int c = 0; c < 4; ++c) {
      const float g = G[q][c];
      const float r = (g == 0.0f) ? NEGF : (g - mn) * inv - 1.0f;
      RES[q][c] = r;
      rmax = fmaxf(rmax, r);
    }
  }

  float ssum = 0.0f;
#pragma unroll
  for (int q = 0; q < 8; ++q) {
#pragma unroll
    for (int c = 0; c < 4; ++c) {
      const float s = __expf(RES[q][c] - rmax);
      RES[q][c] = s;
      ssum += s;
    }
  }

  const float k = sig / ssum;
  f4* o4 = (f4*)(out + pix * RR);
#pragma unroll
  for (int q = 0; q < 8; ++q) {
    __builtin_nontemporal_store(RES[q] * M[q] * k, o4 + q);
  }
}

// ---------------- launch ----------------
extern "C" void kernel_launch(void* const* d_in, const int* in_sizes, int n_in,
                              void* d_out, int out_size, void* d_ws, size_t ws_size,
                              hipStream_t stream) {
  (void)in_sizes; (void)n_in; (void)out_size; (void)ws_size;
  const float* rs   = (const float*)d_in[0];  // relation_stocks      [N,N,D]
  const float* grad = (const float*)d_in[1];  // grad                 [N,N,D]
  const float* rm   = (const float*)d_in[2];  // relation_matrix      [N,N,R]
  const float* rmg  = (const float*)d_in[3];  // relation_matrix_grad [N,N,R]
  float* out  = (float*)d_out;                // [N,N,R]

  float* ewm  = (float*)d_ws;                 // N*N floats (4 MB)
  float* acc  = ewm + (size_t)NN * NN;        // [cnt, sum, sumsq, pad]
  float* stat = acc + 4;                      // [mean, 1/std]

  k0_init<<<1, 32, 0, stream>>>(acc);

  const size_t smem_bytes = (size_t)SMEM_FLOATS * sizeof(float);  // ~135 KB < 320 KB WGP
  k1_ewm_wmma<<<dim3(64 * 64), 256, smem_bytes, stream>>>(rs, grad, ewm, acc);

  k2_stats<<<1, 1, 0, stream>>>(acc, stat);

  k3_softmax_scale<<<dim3((NN * NN) / 256), 256, 0, stream>>>(rm, rmg, ewm, stat, out);
}